// DetectionLoss_89575837925747
// MI455X (gfx1250) — compile-verified
//
#include <hip/hip_runtime.h>
#include <hip/hip_bf16.h>

// DetectionLoss (YOLO-style) fused single-pass reduction for MI455X (gfx1250).
// Memory-bound: ~185 MB streamed -> ~8us floor @ 23.3 TB/s. One thread per
// (b,i,j) cell; per-prior losses computed on the fly and selected by running
// argmax; wave-level reduction uses V_WMMA_F32_16X16X4_F32 (A=ones trick,
// layout-independent exact f32 sum) + shfl_xor, then LDS + global atomic add.

#define S_DIM 13
#define NP 5
#define NC 20
#define E_DIM (5 + NC)
#define SS (S_DIM * S_DIM)
#define LAMBDA_OBJ 5.0f

typedef __attribute__((ext_vector_type(2))) float v2f;
typedef __attribute__((ext_vector_type(8))) float v8f;

__device__ __forceinline__ float wave32_reduce_sum(float acc) {
#if __has_builtin(__builtin_amdgcn_wmma_f32_16x16x4_f32)
    // A = ones(16x4)  -> every D[m][n] = sum_k B[k][n] (column sums).
    // B slots hold the 32 per-lane partials (+32 zeros); since we sum ALL
    // columns afterwards, the internal B striping is irrelevant:
    //   sum_{n} colsum[n] = sum of all 64 B entries = wave total.
    v2f a; a[0] = 1.0f; a[1] = 1.0f;
    v2f b; b[0] = acc;  b[1] = 0.0f;
    v8f c = {};
    v8f d = __builtin_amdgcn_wmma_f32_16x16x4_f32(
        /*neg_a=*/false, a, /*neg_b=*/false, b,
        /*c_mod=*/(short)0, c, /*reuse_a=*/false, /*reuse_b=*/false);
    // D VGPR0: lanes 0-15 = D[0][lane] = colsum[lane];
    //          lanes 16-31 = D[8][lane-16] = colsum[lane-16] (duplicate).
    float col = d[0];
    #pragma unroll
    for (int off = 8; off >= 1; off >>= 1)
        col += __shfl_xor(col, off, 32);   // reduce within each 16-lane group
    return col;                            // every lane now holds the wave sum
#else
    #pragma unroll
    for (int off = 16; off >= 1; off >>= 1)
        acc += __shfl_xor(acc, off, 32);
    return acc;
#endif
}

__global__ void detloss_zero_kernel(float* __restrict__ out) {
    if (threadIdx.x == 0 && blockIdx.x == 0) out[0] = 0.0f;
}

__global__ void __launch_bounds__(256)
DetectionLoss_89575837925747_kernel(const float* __restrict__ pred,
                                    const float* __restrict__ yhat,
                                    const float* __restrict__ priors,
                                    const int*   __restrict__ p_iw,
                                    const int*   __restrict__ p_ih,
                                    float* __restrict__ out,
                                    int Bn, int ncells)
{
    const int t = blockIdx.x * blockDim.x + threadIdx.x;

    const float image_w = (float)(*p_iw);
    const float image_h = (float)(*p_ih);
    const float dx = image_w / (float)S_DIM;   // 32.0 for 416/13
    const float dy = image_h / (float)S_DIM;
    const float invB = 1.0f / (float)Bn;

    float partial = 0.0f;
    if (t < ncells) {
        const int cell = t % SS;
        const int b    = t / SS;
        const int j    = cell % S_DIM;   // x index (fastest-varying)
        const int i    = cell / S_DIM;   // y index

        // y_hat[b,i,j,0:6] contiguous
        const float* yh = yhat + (size_t)t * 6;
        const float y_t0 = yh[0];
        const float g_tx = yh[1], g_ty = yh[2], g_tw = yh[3], g_th = yh[4];
        const float gclsf = yh[5];

        // Ground-truth box in pixel space (exact trunc/floor per reference)
        const float gw  = truncf(g_tw * image_w);
        const float gh  = truncf(g_th * image_h);
        const float gcx = dx * (float)j + truncf(dx * g_tx);
        const float gcy = dy * (float)i + truncf(dy * g_ty);
        const float gx1 = gcx - floorf(gw * 0.5f);
        const float gy1 = gcy - floorf(gh * 0.5f);
        const float gx2 = gx1 + gw;
        const float gy2 = gy1 + gh;

        int gidx = ((int)gclsf - 1) % NC;
        if (gidx < 0) gidx += NC;

        // pred[b, p*E+e, i, j] = pred[(b*NP*E + p*E + e)*169 + cell]
        const size_t estr = (size_t)SS;
        const float* pb = pred + (size_t)b * (NP * E_DIM) * SS + cell;

        float sq_cls    = 0.0f;   // sum over ALL priors of class^2 (noobj term)
        float best_iou  = -1.0f;  // strict > keeps first-max (jnp.argmax)
        float best_cand = 0.0f;

        for (int p = 0; p < NP; ++p) {
            const float* pp = pb + (size_t)(p * E_DIM) * estr;
            const float pr0  = pp[0];
            const float p_tx = pp[1 * estr];
            const float p_ty = pp[2 * estr];
            const float p_tw = pp[3 * estr];
            const float p_th = pp[4 * estr];

            // class terms: accumulate both sum(c^2) and sum((c-onehot)^2)
            float sq_p = 0.0f, cls_p = 0.0f;
            #pragma unroll
            for (int c = 0; c < NC; ++c) {
                const float v = pp[(size_t)(5 + c) * estr];
                sq_p += v * v;
                const float dvc = v - ((c == gidx) ? 1.0f : 0.0f);
                cls_p += dvc * dvc;
            }
            sq_cls += sq_p;

            // predicted box in pixel space
            const float pw  = truncf(priors[2 * p] * p_tw * image_w);
            const float ph  = truncf(priors[2 * p + 1] * p_th * image_h);
            const float pcx = dx * (float)j + truncf(dx * p_tx);
            const float pcy = dy * (float)i + truncf(dy * p_ty);
            const float px1 = pcx - floorf(pw * 0.5f);
            const float py1 = pcy - floorf(ph * 0.5f);
            const float px2 = px1 + pw;
            const float py2 = py1 + ph;

            // IoU (reference semantics: union > 0 guard)
            const float ix1 = fmaxf(px1, gx1);
            const float iy1 = fmaxf(py1, gy1);
            const float ix2 = fminf(px2, gx2);
            const float iy2 = fminf(py2, gy2);
            const float inter  = fmaxf(ix2 - ix1, 0.0f) * fmaxf(iy2 - iy1, 0.0f);
            const float area_p = (px2 - px1) * (py2 - py1);
            const float area_g = (gx2 - gx1) * (gy2 - gy1);
            const float uni    = area_p + area_g - inter;
            const float iou    = (uni > 0.0f) ? (inter / uni) : 0.0f;

            // candidate objective for this prior (valid if it is the argmax,
            // since obj_loss uses max_iou == iou[best])
            const float d0 = p_tx - g_tx, d1 = p_ty - g_ty;
            const float d2 = p_tw - g_tw, d3 = p_th - g_th;
            const float box_p = LAMBDA_OBJ * (d0 * d0 + d1 * d1 + d2 * d2 + d3 * d3);
            const float o     = pr0 * iou - y_t0;
            const float cand  = box_p + o * o + cls_p;

            if (iou > best_iou) { best_iou = iou; best_cand = cand; }
        }

        const float mask = ((y_t0 == 1.0f) && (best_iou >= 0.5f)) ? 1.0f : 0.0f;
        partial = ((1.0f - y_t0) * sq_cls + mask * best_cand) * invB;
    }
    // ---- reconverged here: EXEC all-ones for the WMMA reduction ----

    const float wsum = wave32_reduce_sum(partial);

    __shared__ float ws[8];                 // 256 threads = 8 wave32
    const int wid  = threadIdx.x >> 5;
    const int lane = threadIdx.x & 31;
    if (lane == 0) ws[wid] = wsum;
    __syncthreads();

    if (threadIdx.x == 0) {
        float s = 0.0f;
        const int nw = (blockDim.x + 31) >> 5;
        for (int w = 0; w < nw; ++w) s += ws[w];
        atomicAdd(out, s);
    }
}

extern "C" void kernel_launch(void* const* d_in, const int* in_sizes, int n_in,
                              void* d_out, int out_size, void* d_ws, size_t ws_size,
                              hipStream_t stream) {
    const float* pred   = (const float*)d_in[0];
    const float* yhat   = (const float*)d_in[1];
    const float* priors = (const float*)d_in[2];
    // d_in[3] = inp : unused by the reference loss
    // d_in[4] = num_classes (hardcoded NC=20 to match reference setup)
    const int* p_iw = (const int*)d_in[5];
    const int* p_ih = (const int*)d_in[6];
    float* out = (float*)d_out;

    const int ncells = in_sizes[1] / 6;        // B * S * S  (y_hat is B,S,S,6)
    const int Bn     = ncells / SS;

    detloss_zero_kernel<<<1, 64, 0, stream>>>(out);

    const int threads = 256;
    const int blocks  = (ncells + threads - 1) / threads;   // 1352 for B=2048
    DetectionLoss_89575837925747_kernel<<<blocks, threads, 0, stream>>>(
        pred, yhat, priors, p_iw, p_ih, out, Bn, ncells);
}